// BandpassFilter_90142773608912
// MI455X (gfx1250) — compile-verified
//
#include <hip/hip_runtime.h>

// ---------------------------------------------------------------------------
// BandpassFilter: two order-1 IIR filters (high-pass + low-pass), summed.
// 3-phase exact blocked scan. Phase 3: TDM (tensor_load_to_lds) stages each
// wave's contiguous 32KB input region into LDS (double-buffered, TENSORcnt
// pipelined), then V_WMMA_F32_16X16X4_F32 evaluates 16-sample tiles as
// lower-triangular Toeplitz matmuls across 16 (row,chunk) columns per wave32.
// ---------------------------------------------------------------------------

#define NYQ_F     22050.0f
#define PI_F      3.14159265358979323846f

#define BATCH     64
#define TLEN      262144
#define NCHUNK    512                    // chunks per row
#define LCHUNK    (TLEN / NCHUNK)        // 512 samples per chunk
#define NITEMS    (BATCH * NCHUNK)       // 32768 (row,chunk) items

#define SEG       128                    // samples per TDM segment
#define NSEG      (LCHUNK / SEG)         // 4 segments per chunk
#define TILESEG   (SEG / 16)             // 8 WMMA tiles per segment
#define COLSTRIDE 132                    // LDS dwords per column (128 + 4 pad)
#define COL_F4    (COLSTRIDE / 4)        // 33 float4 per column
#define BUF_F4    (16 * COL_F4)          // 528 float4 per buffer (one segment)
#define WPB       2                      // waves per block in phase 3

typedef __attribute__((ext_vector_type(2))) float v2f;
typedef __attribute__((ext_vector_type(8))) float v8f;
typedef unsigned int u32x4 __attribute__((ext_vector_type(4)));
typedef int          i32x8 __attribute__((ext_vector_type(8)));
typedef int          i32x4 __attribute__((ext_vector_type(4)));

struct Coeffs { float bh0, ah1, bl0, al1; };

// butter(1, wn) via bilinear transform; high-pass b1 = -b0, low-pass b1 = +b0.
__device__ __forceinline__ Coeffs make_coeffs(float lc_in, float hc_in) {
  float low  = fminf(fmaxf(lc_in, 0.0f), NYQ_F);
  float high = fminf(fmaxf(hc_in, low), NYQ_F);
  float th = tanf(PI_F * low  / (2.0f * NYQ_F));   // tan(pi*wn/2), wn = low/nyq
  float tl = tanf(PI_F * high / (2.0f * NYQ_F));
  Coeffs c;
  c.ah1 = (th - 1.0f) / (th + 1.0f);
  c.bh0 = 1.0f / (1.0f + th);            // high-pass: f = bh0*(x[n]-x[n-1])
  c.al1 = (tl - 1.0f) / (tl + 1.0f);
  c.bl0 = tl / (1.0f + tl);              // low-pass:  f = bl0*(x[n]+x[n-1])
  return c;
}

// r^e for e in [0,31]: branch-free binary-exponentiation ladder.
__device__ __forceinline__ float pow_ladder(float r, int e) {
  float r2 = r * r, r4 = r2 * r2, r8 = r4 * r4, r16 = r8 * r8;
  float p = (e & 1)  ? r   : 1.0f;
  p      *= (e & 2)  ? r2  : 1.0f;
  p      *= (e & 4)  ? r4  : 1.0f;
  p      *= (e & 8)  ? r8  : 1.0f;
  p      *= (e & 16) ? r16 : 1.0f;
  return p;
}

__device__ __forceinline__ float lane_bcast(float v, int src_lane) {
  return __int_as_float(__builtin_amdgcn_ds_bpermute(src_lane << 2, __float_as_int(v)));
}

// ---------------------------------------------------------------------------
// TDM: async 2D tile load, 16 columns x SEG samples, into LDS with padding.
// D# per ISA 8.3/8.4: count=1, type=2(image), data_size=2(4B),
// pad_enable, pad_interval=6 (128 dwords), pad_amount=3 (4 dwords)
// -> LDS column stride = 132 dwords (conflict-free ds_load_b128: bank 4*lane).
// tensor_dim0=tile_dim0=SEG, tensor_dim1=tile_dim1=16, dim0_stride=LCHUNK.
// 6-arg clang-23 builtin: (u32x4, i32x8, i32x4, i32x4, i32x8, i32 cpol);
// groups 2/3 + extra group unused for a 2D tensor -> zero-filled.
// ---------------------------------------------------------------------------
__device__ __forceinline__ void tdm_load_seg(const float* gptr, unsigned lds_byte) {
  unsigned long long ga = (unsigned long long)(uintptr_t)gptr;
  u32x4 g0;
  g0[0] = 1u;                                              // count=1, user mode
  g0[1] = lds_byte;                                        // lds_addr
  g0[2] = (unsigned)ga;                                    // global_addr[31:0]
  g0[3] = (unsigned)((ga >> 32) & 0x01FFFFFFu) | (2u << 30); // addr[56:32], type=2
  i32x8 g1;
  g1[0] = (2 << 16) | (1 << 20) | (6 << 22) | (3 << 25);   // dsz=4B, pad 4dw/128dw
  g1[1] = (int)((unsigned)SEG << 16);                      // tensor_dim0[15:0]
  g1[2] = (int)(16u << 16);                                // tensor_dim1 = 16
  g1[3] = (int)((unsigned)SEG << 16);                      // tile_dim0 = SEG
  g1[4] = 16;                                              // tile_dim1 = 16
  g1[5] = LCHUNK;                                          // tensor_dim0_stride
  g1[6] = 0;
  g1[7] = 0;
  i32x4 gz4 = {0, 0, 0, 0};
  i32x8 gz8 = {0, 0, 0, 0, 0, 0, 0, 0};
  __builtin_amdgcn_tensor_load_to_lds(g0, g1, gz4, gz4, gz8, 0);
}

// ---------------------------------------------------------------------------
// Phase 1: per-(row,chunk) zero-init carries z = chunk-local final y.
// ---------------------------------------------------------------------------
__global__ void bp_carry(const float* __restrict__ x,
                         const float* __restrict__ lc, const float* __restrict__ hc,
                         float* __restrict__ zh, float* __restrict__ zl) {
  int item  = blockIdx.x * blockDim.x + threadIdx.x;   // grid sized exactly
  Coeffs cf = make_coeffs(*lc, *hc);
  int  chunk = item & (NCHUNK - 1);
  long base  = (long)item * LCHUNK;                    // == row*TLEN + chunk*LCHUNK
  float xp = (chunk == 0) ? 0.0f : x[base - 1];
  float yh = 0.0f, yl = 0.0f;
#pragma unroll 2
  for (int n = 0; n < LCHUNK; n += 4) {
    float4 q = *(const float4*)(x + base + n);
    yh = cf.bh0 * (q.x - xp) - cf.ah1 * yh;  yl = cf.bl0 * (q.x + xp) - cf.al1 * yl;
    yh = cf.bh0 * (q.y - q.x) - cf.ah1 * yh; yl = cf.bl0 * (q.y + q.x) - cf.al1 * yl;
    yh = cf.bh0 * (q.z - q.y) - cf.ah1 * yh; yl = cf.bl0 * (q.z + q.y) - cf.al1 * yl;
    yh = cf.bh0 * (q.w - q.z) - cf.ah1 * yh; yl = cf.bl0 * (q.w + q.z) - cf.al1 * yl;
    xp = q.w;
  }
  zh[item] = yh;
  zl[item] = yl;
}

// ---------------------------------------------------------------------------
// Phase 2: cross-chunk scan per row: Y_k = r^L * Y_{k-1} + z_k.
// ---------------------------------------------------------------------------
__global__ void bp_scan(const float* __restrict__ lc, const float* __restrict__ hc,
                        const float* __restrict__ zh, const float* __restrict__ zl,
                        float* __restrict__ ih, float* __restrict__ il) {
  int row = blockIdx.x * blockDim.x + threadIdx.x;
  if (row >= BATCH) return;
  Coeffs cf = make_coeffs(*lc, *hc);
  float rh = -cf.ah1, rl = -cf.al1;
  float Ah = rh, Al = rl;
#pragma unroll
  for (int i = 0; i < 9; ++i) { Ah *= Ah; Al *= Al; }  // r^(2^9) = r^LCHUNK
  float Yh = 0.0f, Yl = 0.0f;
#pragma unroll 1
  for (int c = 0; c < NCHUNK; ++c) {
    int idx = row * NCHUNK + c;
    ih[idx] = Yh;                         // initial y entering chunk c
    il[idx] = Yl;
    Yh = Ah * Yh + zh[idx];
    Yl = Al * Yl + zl[idx];
  }
}

// ---------------------------------------------------------------------------
// Phase 3: TDM-staged WMMA tile solve. Per 16-sample tile:
// Y = T*F + p*y_prev, T[i][j] = r^(i-j) (i>=j), p_i = r^(i+1); T*F as 4
// chained v_wmma_f32_16x16x4_f32 (K=16 split 4x4).
// f32 operand layouts (ISA 7.12.2, wave32):
//   A 16x4 : VGPR v, half h -> K = v + 2h, M = lane&15
//   B  4x16: VGPR v, half h -> K = v + 2h, N = lane&15
//   C/D    : VGPR i, half h -> M = i + 8h, N = lane&15
// B operands are built from float4 components only (no 16-element arrays!) to
// keep SLP from forming <16 x float> vectors whose variable-index extracts
// lower to giant v_cmp/v_cndmask mux trees + SGPR spills.
// ---------------------------------------------------------------------------
__global__ void bp_apply(const float* __restrict__ x,
                         const float* __restrict__ lc, const float* __restrict__ hc,
                         const float* __restrict__ ih, const float* __restrict__ il,
                         float* __restrict__ out) {
  __shared__ float4 lds_x[WPB * 2 * BUF_F4];           // 33792 B, 16B aligned

  const int lane = threadIdx.x & 31;
  const int l15  = lane & 15;
  const int h    = lane >> 4;                          // lane half (0 or 1)
  const bool hh  = (h != 0);
  const int wib  = threadIdx.x >> 5;                   // wave in block
  const int wave = (blockIdx.x * blockDim.x + threadIdx.x) >> 5;
  const int item = wave * 16 + l15;                    // this lane's column
  const int chunk = item & (NCHUNK - 1);
  const long base = (long)item * LCHUNK;               // == row*TLEN + chunk*LCHUNK

  Coeffs cf = make_coeffs(*lc, *hc);
  const float rh = -cf.ah1, rl = -cf.al1;

  // Toeplitz A matrices: sub-tile t covers columns j = 4t..4t+3 of T.
  v2f Ath[4], Atl[4];
#pragma unroll
  for (int t = 0; t < 4; ++t) {
#pragma unroll
    for (int v = 0; v < 2; ++v) {
      int j = 4 * t + v + 2 * h;
      int e = l15 - j;                                 // per-lane exponent
      Ath[t][v] = (e >= 0) ? pow_ladder(rh, e & 31) : 0.0f;
      Atl[t][v] = (e >= 0) ? pow_ladder(rl, e & 31) : 0.0f;
    }
  }
  // p vectors: accumulator VGPR i holds row M = i + 8h -> p = r^(M+1)
  float ph[8], pl[8];
#pragma unroll
  for (int i = 0; i < 8; ++i) {
    ph[i] = pow_ladder(rh, i + 8 * h + 1);
    pl[i] = pow_ladder(rl, i + 8 * h + 1);
  }

  float yh_prev = ih[item];
  float yl_prev = il[item];
  float xp = (chunk == 0) ? 0.0f : x[base - 1];

  const float* wave_g = x + (long)wave * 16 * LCHUNK;  // contiguous 32KB region
  float4* wave_l = lds_x + wib * (2 * BUF_F4);

  // Prime the pipeline: segment 0 -> buffer 0.
  tdm_load_seg(wave_g, (unsigned)(uintptr_t)wave_l);

#pragma unroll 1
  for (int seg = 0; seg < NSEG; ++seg) {
    if (seg + 1 < NSEG) {
      // Guard WAR on the buffer being refilled, then issue next segment.
      __asm__ volatile("s_wait_dscnt 0x0" ::: "memory");
      tdm_load_seg(wave_g + (long)(seg + 1) * SEG,
                   (unsigned)(uintptr_t)(wave_l + ((seg + 1) & 1) * BUF_F4));
      __builtin_amdgcn_s_wait_tensorcnt(1);  // TDM in-order: current buf ready
    } else {
      __builtin_amdgcn_s_wait_tensorcnt(0);
    }
    __asm__ volatile("" ::: "memory");       // LDS now holds TDM data

    const float4* colp = wave_l + ((seg & 1) * BUF_F4) + l15 * COL_F4;

#pragma unroll 2
    for (int s = 0; s < TILESEG; ++s) {
      const long t0 = base + (long)seg * SEG + (long)s * 16;

      float4 q0 = colp[s * 4 + 0];                     // ds_load_b128 x4
      float4 q1 = colp[s * 4 + 1];
      float4 q2 = colp[s * 4 + 2];
      float4 q3 = colp[s * 4 + 3];

      // Accumulators start at p * y_prev (state propagation term).
      v8f ch, clw;
#pragma unroll
      for (int i = 0; i < 8; ++i) { ch[i] = ph[i] * yh_prev; clw[i] = pl[i] * yl_prev; }

      // Sub-tile t: lane half h supplies B rows K = 4t+2h, 4t+2h+1, i.e.
      // samples xa = x[4t+2h], xb = x[4t+2h+1], predecessor xm = x[4t+2h-1].
      // All picked from float4 components -> no vectorizable 16-array.
#define SUBTILE(T, QT, QPREV_W)                                               \
      {                                                                       \
        float xa = hh ? (QT).z : (QT).x;                                      \
        float xb = hh ? (QT).w : (QT).y;                                      \
        float xm = hh ? (QT).y : (QPREV_W);                                   \
        v2f bh, bl;                                                           \
        bh[0] = cf.bh0 * (xa - xm); bh[1] = cf.bh0 * (xb - xa);               \
        bl[0] = cf.bl0 * (xa + xm); bl[1] = cf.bl0 * (xb + xa);               \
        ch  = __builtin_amdgcn_wmma_f32_16x16x4_f32(false, Ath[T], false, bh, \
                                                    (short)0, ch,  false, false); \
        clw = __builtin_amdgcn_wmma_f32_16x16x4_f32(false, Atl[T], false, bl, \
                                                    (short)0, clw, false, false); \
      }
      SUBTILE(0, q0, xp)
      SUBTILE(1, q1, q0.w)
      SUBTILE(2, q2, q1.w)
      SUBTILE(3, q3, q2.w)
#undef SUBTILE
      xp = q3.w;

      // Next tile's state: row M=15 lives in accumulator VGPR 7, lanes 16..31.
      yh_prev = lane_bcast(ch[7],  16 + l15);
      yl_prev = lane_bcast(clw[7], 16 + l15);

      // Store: this lane holds times t0 + 8h + (0..7) of its column.
      const long ob = t0 + 8 * h;
      float4 o0 = make_float4(ch[0] + clw[0], ch[1] + clw[1],
                              ch[2] + clw[2], ch[3] + clw[3]);
      float4 o1 = make_float4(ch[4] + clw[4], ch[5] + clw[5],
                              ch[6] + clw[6], ch[7] + clw[7]);
      *(float4*)(out + ob)     = o0;
      *(float4*)(out + ob + 4) = o1;
    }
  }
}

// ---------------------------------------------------------------------------
extern "C" void kernel_launch(void* const* d_in, const int* in_sizes, int n_in,
                              void* d_out, int out_size, void* d_ws, size_t ws_size,
                              hipStream_t stream) {
  const float* x  = (const float*)d_in[0];
  const float* lc = (const float*)d_in[1];
  const float* hc = (const float*)d_in[2];
  float* out = (float*)d_out;

  float* zh = (float*)d_ws;              // 4 * NITEMS floats = 512 KB total
  float* zl = zh + NITEMS;
  float* ih = zl + NITEMS;
  float* il = ih + NITEMS;

  // Phase 1: chunk carries (exact grid, no bounds checks needed)
  bp_carry<<<NITEMS / 256, 256, 0, stream>>>(x, lc, hc, zh, zl);
  // Phase 2: cross-chunk scan per row
  bp_scan<<<1, 64, 0, stream>>>(lc, hc, zh, zl, ih, il);
  // Phase 3: TDM + WMMA tile solve; 2048 full waves, 2 waves (64 thr) / block
  bp_apply<<<(NITEMS / 16) / WPB, WPB * 32, 0, stream>>>(x, lc, hc, ih, il, out);
}